// MultiheadedAttention_15152644620499
// MI455X (gfx1250) — compile-verified
//
#include <hip/hip_runtime.h>
#include <math.h>

// Problem dims
#define Bn 4
#define Cn 512
#define Ln 2048
#define An 512
#define Hn 8
#define Dn 64

typedef __attribute__((ext_vector_type(2))) float v2f;
typedef __attribute__((ext_vector_type(8))) float v8f;

// D = A(16x4 f32) * B(4x16 f32) + C(16x16 f32), wave32
static __device__ __forceinline__ v8f wmma_f32(v2f a, v2f b, v8f c) {
  return __builtin_amdgcn_wmma_f32_16x16x4_f32(false, a, false, b, (short)0, c,
                                               false, false);
}

// Per-lane async copy of 16B global -> LDS, tracked by ASYNCcnt.
static __device__ __forceinline__ void async_load_b128(unsigned lds_byte,
                                                       const float* gaddr) {
  asm volatile("global_load_async_to_lds_b128 %0, %1, off" ::"v"(lds_byte),
               "v"(gaddr)
               : "memory");
}
static __device__ __forceinline__ void wait_async0() {
  asm volatile("s_wait_asynccnt 0x0" ::: "memory");
}

static __device__ __forceinline__ void atomicMaxF(float* addr, float val) {
  if (val >= 0.0f)
    atomicMax((int*)addr, __float_as_int(val));
  else
    atomicMin((unsigned int*)addr, __float_as_uint(val));
}

__global__ void init_stats(float* maxbuf, float* sumbuf) {
  int i = threadIdx.x;
  if (i < 32) {
    maxbuf[i] = -INFINITY;
    sumbuf[i] = 0.0f;
  }
}

// ---------------------------------------------------------------------------
// Kernel 1: q/k/v projections.  32(d) x 32(l) wave tile; x B-frags shared by
// q/k/v -> 12 wmma per K-step off 2 B-frag + 6 A-frag loads.  K = C = 512.
// ---------------------------------------------------------------------------
__global__ __launch_bounds__(256) void proj_qkv(
    const float* __restrict__ x, const float* __restrict__ wk,
    const float* __restrict__ bk, const float* __restrict__ wq,
    const float* __restrict__ bq, const float* __restrict__ wv,
    const float* __restrict__ bv, float* __restrict__ qb,
    float* __restrict__ kb, float* __restrict__ vb) {
  int wid = blockIdx.x * 8 + (threadIdx.x >> 5);
  int lane = threadIdx.x & 31;
  int lt = wid & 63;         // 64 l-tiles of 32
  int dt = (wid >> 6) & 1;   // 2 d-tiles of 32
  int h = (wid >> 7) & 7;
  int b = wid >> 10;
  int l0 = lt * 32, d0 = dt * 32;
  int col = lane & 15;
  int kh = lane >> 4;

  const float* xb = x + (size_t)b * Cn * Ln;
  int n[2], wrow[2];
#pragma unroll
  for (int s = 0; s < 2; ++s) {
    n[s] = l0 + s * 16 + col;
    wrow[s] = (h * Dn + d0 + s * 16 + col) * Cn;
  }
  v8f cq[2][2] = {{{}, {}}, {{}, {}}};
  v8f ck[2][2] = {{{}, {}}, {{}, {}}};
  v8f cv[2][2] = {{{}, {}}, {{}, {}}};

  for (int c0 = 0; c0 < Cn; c0 += 4) {
    int cc = c0 + 2 * kh;
    v2f bx[2], aq[2], ak[2], av[2];
#pragma unroll
    for (int s = 0; s < 2; ++s) {
      bx[s].x = xb[cc * Ln + n[s]];
      bx[s].y = xb[(cc + 1) * Ln + n[s]];
      aq[s].x = wq[wrow[s] + cc];
      aq[s].y = wq[wrow[s] + cc + 1];
      ak[s].x = wk[wrow[s] + cc];
      ak[s].y = wk[wrow[s] + cc + 1];
      av[s].x = wv[wrow[s] + cc];
      av[s].y = wv[wrow[s] + cc + 1];
    }
#pragma unroll
    for (int ms = 0; ms < 2; ++ms)
#pragma unroll
      for (int ns = 0; ns < 2; ++ns) {
        cq[ms][ns] = wmma_f32(aq[ms], bx[ns], cq[ms][ns]);
        ck[ms][ns] = wmma_f32(ak[ms], bx[ns], ck[ms][ns]);
        cv[ms][ns] = wmma_f32(av[ms], bx[ns], cv[ms][ns]);
      }
  }
  int bh = b * Hn + h;
#pragma unroll
  for (int ms = 0; ms < 2; ++ms)
#pragma unroll
    for (int r = 0; r < 8; ++r) {
      int d = d0 + ms * 16 + r + 8 * kh;
      float biq = bq[h * Dn + d], bik = bk[h * Dn + d], biv = bv[h * Dn + d];
#pragma unroll
      for (int ns = 0; ns < 2; ++ns) {
        size_t o = ((size_t)bh * Dn + d) * Ln + n[ns];
        qb[o] = cq[ms][ns][r] + biq;
        kb[o] = ck[ms][ns][r] + bik;
        vb[o] = cv[ms][ns][r] + biv;
      }
    }
}

// ---------------------------------------------------------------------------
// Kernel 2: global max of (K^T Q)/8 per (b,h).  32x32 score tiles; 256 waves
// per (b,h), each covering a 16-tile l strip for one 32-wide m tile.
// ---------------------------------------------------------------------------
__global__ __launch_bounds__(256) void score_max(const float* __restrict__ qb,
                                                 const float* __restrict__ kb,
                                                 float* __restrict__ maxbuf) {
  int wid = blockIdx.x * 8 + (threadIdx.x >> 5);
  int lane = threadIdx.x & 31;
  int wl = wid & 255;
  int bh = wid >> 8;
  int mt = wl & 63;       // 64 m-tiles of 32
  int strip = wl >> 6;    // 4 strips of 16 l-tiles
  int col = lane & 15, kh = lane >> 4;
  const float* kbase = kb + (size_t)bh * Dn * Ln;
  const float* qbase = qb + (size_t)bh * Dn * Ln;
  int m0 = mt * 32;
  int mcol[2] = {m0 + col, m0 + 16 + col};
  float lmax = -INFINITY;
  for (int lt = strip * 16; lt < strip * 16 + 16; ++lt) {
    int l0 = lt * 32;
    int lrow[2] = {l0 + col, l0 + 16 + col};
    v8f s[2][2] = {{{}, {}}, {{}, {}}};
    for (int kc = 0; kc < Dn; kc += 4) {
      int d = kc + 2 * kh;
      v2f a[2], bf[2];
#pragma unroll
      for (int t = 0; t < 2; ++t) {
        a[t].x = kbase[d * Ln + lrow[t]];
        a[t].y = kbase[(d + 1) * Ln + lrow[t]];
        bf[t].x = qbase[d * Ln + mcol[t]];
        bf[t].y = qbase[(d + 1) * Ln + mcol[t]];
      }
#pragma unroll
      for (int ms = 0; ms < 2; ++ms)
#pragma unroll
        for (int ns = 0; ns < 2; ++ns) s[ms][ns] = wmma_f32(a[ms], bf[ns], s[ms][ns]);
    }
#pragma unroll
    for (int ms = 0; ms < 2; ++ms)
#pragma unroll
      for (int ns = 0; ns < 2; ++ns)
#pragma unroll
        for (int r = 0; r < 8; ++r) lmax = fmaxf(lmax, s[ms][ns][r] * 0.125f);
  }
  for (int off = 16; off > 0; off >>= 1)
    lmax = fmaxf(lmax, __shfl_xor(lmax, off, 32));
  if (lane == 0) atomicMaxF(maxbuf + bh, lmax);
}

// ---------------------------------------------------------------------------
// Kernel 3: fused exp + PV.  Per wave: 32-wide m tile; loop over 64 l-tiles
// of 32: recompute 32x32 scores, p = exp(s/8 - gmax) through LDS into WMMA
// B-layout, accumulate sa_unnorm[64 x 32].  Atomic sum of exp per (b,h).
// ---------------------------------------------------------------------------
__global__ __launch_bounds__(128) void attn_pv(
    const float* __restrict__ qb, const float* __restrict__ kb,
    const float* __restrict__ vb, const float* __restrict__ maxbuf,
    float* __restrict__ sa, float* __restrict__ sumbuf) {
  __shared__ float pbuf[4][32][34];  // [wave][row l][col m], padded
  int wslot = threadIdx.x >> 5;
  int wid = blockIdx.x * 4 + wslot;
  int lane = threadIdx.x & 31;
  int mt = wid & 63;
  int bh = wid >> 6;
  int col = lane & 15, kh = lane >> 4;
  const float* kbase = kb + (size_t)bh * Dn * Ln;
  const float* qbase = qb + (size_t)bh * Dn * Ln;
  const float* vbase = vb + (size_t)bh * Dn * Ln;
  float gmax = maxbuf[bh];
  int m0 = mt * 32;
  int mcol[2] = {m0 + col, m0 + 16 + col};
  v8f acc[4][2] = {{{}, {}}, {{}, {}}, {{}, {}}, {{}, {}}};
  float lsum = 0.0f;
  float(*pt)[34] = pbuf[wslot];

  for (int lt = 0; lt < 64; ++lt) {
    int l0 = lt * 32;
    int lrow[2] = {l0 + col, l0 + 16 + col};
    v8f s[2][2] = {{{}, {}}, {{}, {}}};
    for (int kc = 0; kc < Dn; kc += 4) {
      int d = kc + 2 * kh;
      v2f a[2], bf[2];
#pragma unroll
      for (int t = 0; t < 2; ++t) {
        a[t].x = kbase[d * Ln + lrow[t]];
        a[t].y = kbase[(d + 1) * Ln + lrow[t]];
        bf[t].x = qbase[d * Ln + mcol[t]];
        bf[t].y = qbase[(d + 1) * Ln + mcol[t]];
      }
#pragma unroll
      for (int ms = 0; ms < 2; ++ms)
#pragma unroll
        for (int ns = 0; ns < 2; ++ns) s[ms][ns] = wmma_f32(a[ms], bf[ns], s[ms][ns]);
    }
    // p = exp(s/8 - gmax) -> LDS (same-wave DS ops are in-order)
#pragma unroll
    for (int ms = 0; ms < 2; ++ms)
#pragma unroll
      for (int ns = 0; ns < 2; ++ns)
#pragma unroll
        for (int r = 0; r < 8; ++r) {
          float p = __expf(s[ms][ns][r] * 0.125f - gmax);
          lsum += p;
          pt[ms * 16 + r + 8 * kh][ns * 16 + col] = p;
        }
    // acc[t][ns] += V[:, l-tile] * P : K = 32
#pragma unroll
    for (int kc = 0; kc < 32; kc += 4) {
      int kk = kc + 2 * kh;
      v2f bp[2];
#pragma unroll
      for (int ns = 0; ns < 2; ++ns) {
        bp[ns].x = pt[kk][ns * 16 + col];
        bp[ns].y = pt[kk + 1][ns * 16 + col];
      }
#pragma unroll
      for (int t = 0; t < 4; ++t) {
        v2f a;
        int dd = t * 16 + col;
        a.x = vbase[dd * Ln + l0 + kk];
        a.y = vbase[dd * Ln + l0 + kk + 1];
#pragma unroll
        for (int ns = 0; ns < 2; ++ns) acc[t][ns] = wmma_f32(a, bp[ns], acc[t][ns]);
      }
    }
  }
  for (int off = 16; off > 0; off >>= 1) lsum += __shfl_xor(lsum, off, 32);
  if (lane == 0) atomicAdd(sumbuf + bh, lsum);
#pragma unroll
  for (int t = 0; t < 4; ++t)
#pragma unroll
    for (int r = 0; r < 8; ++r) {
      int d = t * 16 + r + 8 * kh;
#pragma unroll
      for (int ns = 0; ns < 2; ++ns)
        sa[((size_t)bh * Dn + d) * Ln + mcol[ns]] = acc[t][ns][r];
    }
}

// ---------------------------------------------------------------------------
// Kernel 4: divide sa_unnorm by the global softmax sum per (b,h).
// ---------------------------------------------------------------------------
__global__ void normalize_sa(float* __restrict__ sa,
                             const float* __restrict__ sumbuf) {
  int idx = blockIdx.x * blockDim.x + threadIdx.x;
  int bh = idx >> 17;  // 64*2048 elements per (b,h)
  sa[idx] = sa[idx] * (1.0f / sumbuf[bh]);
}

// ---------------------------------------------------------------------------
// Kernel 5/6: out = W[512x512]*in[512x2048] + bias (optional ReLU).
// Block = 4 waves sharing one 32-row W panel, async-staged into LDS in 32x64
// chunks via global_load_async_to_lds_b128 (ASYNCcnt), A-frags from LDS,
// B-frags from (L2-resident) input.  32x32 tile per wave.
// ---------------------------------------------------------------------------
__global__ __launch_bounds__(128) void mm512(const float* __restrict__ W,
                                             const float* __restrict__ bias,
                                             const float* __restrict__ in,
                                             float* __restrict__ out,
                                             int relu) {
  __shared__ float Wp[32][68];  // 68-float row stride: 16B-aligned + bank-spread
  int ltg = blockIdx.x & 15;
  int ot = (blockIdx.x >> 4) & 15;
  int b = blockIdx.x >> 8;
  int wslot = threadIdx.x >> 5;
  int lane = threadIdx.x & 31;
  int col = lane & 15, kh = lane >> 4;
  int o0 = ot * 32;
  int l0 = (ltg * 4 + wslot) * 32;
  int n[2] = {l0 + col, l0 + 16 + col};
  const float* inb = in + (size_t)b * An * Ln;
  v8f c[2][2] = {{{}, {}}, {{}, {}}};

  unsigned wp_base = (unsigned)(unsigned long long)(&Wp[0][0]);
  for (int c0 = 0; c0 < An; c0 += 64) {
    // Cooperative async stage of W[o0..o0+32)[c0..c0+64)
#pragma unroll
    for (int i = 0; i < 4; ++i) {
      int id = threadIdx.x + 128 * i;  // 0..511
      int row = id >> 4;
      int cw = (id & 15) << 2;
      async_load_b128(wp_base + (unsigned)(row * 68 + cw) * 4u,
                      W + (o0 + row) * An + c0 + cw);
    }
    wait_async0();
    __syncthreads();
#pragma unroll 4
    for (int kc = 0; kc < 64; kc += 4) {
      int cc = kc + 2 * kh;
      v2f a[2], bf[2];
#pragma unroll
      for (int t = 0; t < 2; ++t) {
        a[t].x = Wp[t * 16 + col][cc];
        a[t].y = Wp[t * 16 + col][cc + 1];
        bf[t].x = inb[(c0 + cc) * Ln + n[t]];
        bf[t].y = inb[(c0 + cc + 1) * Ln + n[t]];
      }
#pragma unroll
      for (int ms = 0; ms < 2; ++ms)
#pragma unroll
        for (int ns = 0; ns < 2; ++ns) c[ms][ns] = wmma_f32(a[ms], bf[ns], c[ms][ns]);
    }
    __syncthreads();
  }
  float* ob = out + (size_t)b * An * Ln;
#pragma unroll
  for (int ms = 0; ms < 2; ++ms)
#pragma unroll
    for (int r = 0; r < 8; ++r) {
      int o = o0 + ms * 16 + r + 8 * kh;
      float bi = bias[o];
#pragma unroll
      for (int ns = 0; ns < 2; ++ns) {
        float v = c[ms][ns][r] + bi;
        if (relu) v = fmaxf(v, 0.0f);
        ob[(size_t)o * Ln + n[ns]] = v;
      }
    }
}

extern "C" void kernel_launch(void* const* d_in, const int* in_sizes, int n_in,
                              void* d_out, int out_size, void* d_ws,
                              size_t ws_size, hipStream_t stream) {
  const float* x = (const float*)d_in[0];
  const float* wk = (const float*)d_in[1];
  const float* bk = (const float*)d_in[2];
  const float* wq = (const float*)d_in[3];
  const float* bq = (const float*)d_in[4];
  const float* wv = (const float*)d_in[5];
  const float* bv = (const float*)d_in[6];
  const float* w1 = (const float*)d_in[7];
  const float* b1 = (const float*)d_in[8];
  const float* w2 = (const float*)d_in[9];
  const float* b2 = (const float*)d_in[10];

  float* ws = (float*)d_ws;
  const size_t QKV = (size_t)Bn * Hn * Dn * Ln;  // 4,194,304 floats
  float* qb = ws;
  float* kb = ws + QKV;
  float* vb = ws + 2 * QKV;
  float* sa = ws + 3 * QKV;
  float* zb = ws + 4 * QKV;
  float* maxbuf = ws + 5 * QKV;
  float* sumbuf = maxbuf + 32;

  init_stats<<<1, 32, 0, stream>>>(maxbuf, sumbuf);
  // 4*8*2*64 = 4096 waves, 8/block
  proj_qkv<<<512, 256, 0, stream>>>(x, wk, bk, wq, bq, wv, bv, qb, kb, vb);
  // 32*256 = 8192 waves
  score_max<<<1024, 256, 0, stream>>>(qb, kb, maxbuf);
  // 32*64 = 2048 waves, 4/block
  attn_pv<<<512, 128, 0, stream>>>(qb, kb, vb, maxbuf, sa, sumbuf);
  // 4,194,304 elements
  normalize_sa<<<16384, 256, 0, stream>>>(sa, sumbuf);
  // 4*16*16 blocks of 4 waves
  mm512<<<1024, 128, 0, stream>>>(w1, b1, sa, zb, 1);
  mm512<<<1024, 128, 0, stream>>>(w2, b2, zb, (float*)d_out, 0);
}